// GCN_classification_weighted_69578470195571
// MI455X (gfx1250) — compile-verified
//
#include <hip/hip_runtime.h>
#include <hip/hip_bf16.h>

#define DFEAT 128
#define KCHUNK 32
#define LDS_STRIDE 36   // 32 + 4 pad: conflict-free for the wmma operand fetch pattern
#define NEG_SLOPE 0.01f

typedef __attribute__((ext_vector_type(2))) float v2f;
typedef __attribute__((ext_vector_type(8))) float v8f;

// ------------------------------------------------------------ CSR build
__global__ void hist_kernel(const int* __restrict__ dst, int* __restrict__ cnt, int E) {
    int e = blockIdx.x * blockDim.x + threadIdx.x;
    if (e < E) atomicAdd(&cnt[dst[e]], 1);
}

// Single-workgroup exclusive scan (N ~ 100k ints: microseconds, deterministic).
__global__ __launch_bounds__(1024) void scan_kernel(const int* __restrict__ cnt,
                                                    int* __restrict__ rowptr, int N) {
    __shared__ int sm[1024];
    __shared__ int carry_s;
    const int tid = threadIdx.x;
    if (tid == 0) carry_s = 0;
    __syncthreads();
    for (int base = 0; base < N; base += 1024) {
        int i = base + tid;
        int v = (i < N) ? cnt[i] : 0;
        sm[tid] = v;
        __syncthreads();
        #pragma unroll
        for (int off = 1; off < 1024; off <<= 1) {
            int t = (tid >= off) ? sm[tid - off] : 0;
            __syncthreads();
            sm[tid] += t;
            __syncthreads();
        }
        int carry = carry_s;
        if (i < N) rowptr[i] = carry + sm[tid] - v;   // exclusive
        __syncthreads();
        if (tid == 1023) carry_s = carry + sm[1023];
        __syncthreads();
    }
    if (tid == 0) rowptr[N] = carry_s;
}

__global__ void csr_fill_kernel(const int* __restrict__ dst, const int* __restrict__ rowptr,
                                int* __restrict__ cursor, int* __restrict__ eid, int E) {
    int e = blockIdx.x * blockDim.x + threadIdx.x;
    if (e < E) {
        int d = dst[e];
        int pos = rowptr[d] + atomicAdd(&cursor[d], 1);
        eid[pos] = e;
    }
}

// ------------------------------------------- pull-mode weighted SpMM
// One wave per destination node. Register accumulation (no float atomics),
// coalesced 512B gathers that stay L2-resident (51MB << 192MB L2). Fuses
// mean (1/deg) and leaky_relu. Software-pipelined edge loop + prefetch.
__global__ void spmm_pull_kernel(const float* __restrict__ x, const int* __restrict__ rowptr,
                                 const int* __restrict__ eid, const int* __restrict__ src,
                                 const float* __restrict__ ew, float* __restrict__ act, int N) {
    int node = (int)((blockIdx.x * blockDim.x + threadIdx.x) >> 5);
    int lane = threadIdx.x & 31;
    if (node >= N) return;
    int beg = rowptr[node];
    int end = rowptr[node + 1];

    float4 acc = {0.0f, 0.0f, 0.0f, 0.0f};
    int   s = 0;
    float w = 0.0f;
    if (beg < end) { int e = eid[beg]; s = src[e]; w = ew[e]; }
    for (int p = beg; p < end; ++p) {
        const float4* xrow = (const float4*)(x + (size_t)s * DFEAT);
        float4 v = xrow[lane];
        float wcur = w;
        if (p + 1 < end) {                       // fetch next edge early
            int e2 = eid[p + 1];
            s = src[e2];
            w = ew[e2];
            __builtin_prefetch((const void*)(x + (size_t)s * DFEAT + lane * 4), 0, 3);
        }
        acc.x = fmaf(wcur, v.x, acc.x);
        acc.y = fmaf(wcur, v.y, acc.y);
        acc.z = fmaf(wcur, v.z, acc.z);
        acc.w = fmaf(wcur, v.w, acc.w);
    }

    int   deg = end - beg;
    float inv = (deg > 0) ? (1.0f / (float)deg) : 0.0f;
    float4 r;
    r.x = acc.x * inv; r.y = acc.y * inv; r.z = acc.z * inv; r.w = acc.w * inv;
    r.x = (r.x > 0.0f) ? r.x : NEG_SLOPE * r.x;
    r.y = (r.y > 0.0f) ? r.y : NEG_SLOPE * r.y;
    r.z = (r.z > 0.0f) ? r.z : NEG_SLOPE * r.z;
    r.w = (r.w > 0.0f) ? r.w : NEG_SLOPE * r.w;
    ((float4*)(act + (size_t)node * DFEAT))[lane] = r;
}

// ------------------------------------------------------- WMMA fp32 GEMM
// C[M x 128] = op( [A0 | A1] [M x NPARTS*128] @ W [NPARTS*128 x 128] + bias )
// 256 threads = 8 waves; block tile 128 rows x 16 cols; K staged through LDS
// 32 wide. Fully unrolled (NPARTS is a template constant). The staged tile
// lives in individually-named scalar registers (no arrays -> no SROA
// failure -> no scratch/LDS spill); after the regs are committed to LDS
// they are dead, so the next chunk's global loads overwrite them in place,
// overlapping HBM/L2 latency with the WMMAs. V_WMMA_F32_16X16X4_F32.
template <int NPARTS>
__global__ __launch_bounds__(256) void gemm_wmma_kernel(
        const float* __restrict__ A0, const float* __restrict__ A1,
        const float* __restrict__ W,  const float* __restrict__ bias,
        float* __restrict__ C, int M, int relu) {

    __shared__ float Alds[128 * LDS_STRIDE];  // [row 0..127][k 0..31]
    __shared__ float Blds[16 * LDS_STRIDE];   // transposed: [n 0..15][k 0..31]

    const int t     = threadIdx.x;
    const int lane  = t & 31;
    const int wave  = t >> 5;
    const int mBase = blockIdx.x * 128;
    const int nBase = blockIdx.y * 16;
    const int lmod  = lane & 15;
    const int lhi   = lane >> 4;

    // per-thread cooperative-load coordinates (constant across chunks)
    const int aq = t & 7;            // quad within row
    const int ar = t >> 3;           // row 0..31 (+32*i)
    const int bn = t & 15;           // B col
    const int bk = t >> 4;           // B row 0..15 (+16)

    int ag0 = mBase + ar;       if (ag0 > M - 1) ag0 = M - 1;
    int ag1 = mBase + ar + 32;  if (ag1 > M - 1) ag1 = M - 1;
    int ag2 = mBase + ar + 64;  if (ag2 > M - 1) ag2 = M - 1;
    int ag3 = mBase + ar + 96;  if (ag3 > M - 1) ag3 = M - 1;

    constexpr int NCHUNK = NPARTS * (DFEAT / KCHUNK);

    // staged tile in named scalars only
    float4 a0, a1, a2, a3;
    float  b0, b1;

#define LOAD_CHUNK(cn)                                                          \
    {                                                                           \
        const int   kb_ = ((cn) & 3) * KCHUNK;                                  \
        const float* __restrict__ Ap_ = (NPARTS == 2 && (cn) >= 4) ? A1 : A0;   \
        const int   wo_ = (NPARTS == 2 && (cn) >= 4) ? DFEAT : 0;               \
        a0 = *(const float4*)(Ap_ + (size_t)ag0 * DFEAT + kb_ + aq * 4);        \
        a1 = *(const float4*)(Ap_ + (size_t)ag1 * DFEAT + kb_ + aq * 4);        \
        a2 = *(const float4*)(Ap_ + (size_t)ag2 * DFEAT + kb_ + aq * 4);        \
        a3 = *(const float4*)(Ap_ + (size_t)ag3 * DFEAT + kb_ + aq * 4);        \
        b0 = W[(size_t)(wo_ + kb_ + bk) * DFEAT + nBase + bn];                  \
        b1 = W[(size_t)(wo_ + kb_ + bk + 16) * DFEAT + nBase + bn];             \
    }

    LOAD_CHUNK(0);

    float4* const sa0 = (float4*)&Alds[(ar +  0) * LDS_STRIDE + aq * 4];
    float4* const sa1 = (float4*)&Alds[(ar + 32) * LDS_STRIDE + aq * 4];
    float4* const sa2 = (float4*)&Alds[(ar + 64) * LDS_STRIDE + aq * 4];
    float4* const sa3 = (float4*)&Alds[(ar + 96) * LDS_STRIDE + aq * 4];
    float*  const sb  = &Blds[bn * LDS_STRIDE + bk];

    const float* aptr = &Alds[(wave * 16 + lmod) * LDS_STRIDE + 2 * lhi];
    const float* bptr = &Blds[lmod * LDS_STRIDE + 2 * lhi];

    v8f acc = {};

    #pragma unroll
    for (int c = 0; c < NCHUNK; ++c) {
        // commit staged registers to LDS
        *sa0 = a0; *sa1 = a1; *sa2 = a2; *sa3 = a3;
        sb[0] = b0; sb[16] = b1;
        __syncthreads();

        // staged regs are dead now: refill them with chunk c+1 so the
        // global loads overlap the WMMAs below
        if (c + 1 < NCHUNK) LOAD_CHUNK(c + 1);

        // a[v] = A[M=lmod][K = kk + v + 2*lhi], b[v] = B[K = kk + v + 2*lhi][N=lmod]
        #pragma unroll
        for (int kk = 0; kk < KCHUNK; kk += 4) {
            v2f a = *(const v2f*)(aptr + kk);
            v2f b = *(const v2f*)(bptr + kk);
            acc = __builtin_amdgcn_wmma_f32_16x16x4_f32(
                      /*neg_a=*/false, a, /*neg_b=*/false, b,
                      /*c_mod=*/(short)0, acc,
                      /*reuse_a=*/false, /*reuse_b=*/false);
        }
        __syncthreads();
    }
#undef LOAD_CHUNK

    // C/D layout: vgpr v, lane l -> row = v + 8*(l/16), col = l%16
    float bvv = bias[nBase + lmod];
    #pragma unroll
    for (int v = 0; v < 8; ++v) {
        int row = mBase + wave * 16 + v + 8 * lhi;
        float val = acc[v] + bvv;
        if (relu) val = fmaxf(val, 0.0f);
        if (row < M) C[(size_t)row * DFEAT + nBase + lmod] = val;
    }
}

// ------------------------------------------------ final [128 x 2] head
__global__ void classifier_kernel(const float* __restrict__ s, const float* __restrict__ Wl2,
                                  const float* __restrict__ b2, float* __restrict__ out, int N) {
    int node = (int)((blockIdx.x * blockDim.x + threadIdx.x) >> 5);
    int lane = threadIdx.x & 31;
    if (node >= N) return;
    const float* row = s + (size_t)node * DFEAT;
    float a0 = 0.0f, a1 = 0.0f;
    #pragma unroll
    for (int i = 0; i < 4; ++i) {
        int d = lane + i * 32;
        float v = row[d];
        a0 += v * Wl2[d * 2 + 0];
        a1 += v * Wl2[d * 2 + 1];
    }
    #pragma unroll
    for (int off = 16; off >= 1; off >>= 1) {
        a0 += __shfl_xor(a0, off, 32);
        a1 += __shfl_xor(a1, off, 32);
    }
    if (lane == 0) {
        out[(size_t)node * 2 + 0] = a0 + b2[0];
        out[(size_t)node * 2 + 1] = a1 + b2[1];
    }
}

// ---------------------------------------------------------------- launch
static inline char* align_up(char* p, size_t a) {
    return (char*)(((uintptr_t)p + (a - 1)) & ~(uintptr_t)(a - 1));
}

extern "C" void kernel_launch(void* const* d_in, const int* in_sizes, int n_in,
                              void* d_out, int out_size, void* d_ws, size_t ws_size,
                              hipStream_t stream) {
    const float* x   = (const float*)d_in[0];
    const int*   esr = (const int*)  d_in[1];
    const int*   eds = (const int*)  d_in[2];
    const float* ew1 = (const float*)d_in[3];
    const float* ew2 = (const float*)d_in[4];
    const float* W1  = (const float*)d_in[5];
    const float* b1  = (const float*)d_in[6];
    const float* W2  = (const float*)d_in[7];
    const float* b2  = (const float*)d_in[8];
    const float* Wl1 = (const float*)d_in[9];
    const float* bl1 = (const float*)d_in[10];
    const float* Wl2 = (const float*)d_in[11];
    const float* bl2 = (const float*)d_in[12];

    const int    N  = in_sizes[0] / DFEAT;   // 100000
    const int    E  = in_sizes[1];           // 1600000
    const size_t ND = (size_t)N * DFEAT;

    char* p = (char*)d_ws;
    int* cnt    = (int*)p;                 p += (size_t)N * 4;
    int* cursor = (int*)p;                 p += (size_t)N * 4;
    int* rowptr = (int*)p;                 p += (size_t)(N + 1) * 4;
    p = align_up(p, 64);
    int* eid    = (int*)p;                 p += (size_t)E * 4;
    p = align_up(p, 64);
    float* act  = (float*)p;               p += ND * 4;   // leaky(mean agg); reused as 's'
    float* hA   = (float*)p;               p += ND * 4;   // layer-1 output
    float* hB   = (float*)p;               p += ND * 4;   // layer-2 output
    float* out  = (float*)d_out;

    const int thr = 256;
    dim3 gemmGrid((N + 127) / 128, DFEAT / 16);
    const int edgeBlocks = (E + thr - 1) / thr;
    const int nodeWaveBlocks = (N * 32 + thr - 1) / thr;

    // ---- CSR build (per call; graph-capture safe)
    hipMemsetAsync(cnt, 0, (size_t)N * 4, stream);
    hipMemsetAsync(cursor, 0, (size_t)N * 4, stream);
    hist_kernel<<<edgeBlocks, thr, 0, stream>>>(eds, cnt, E);
    scan_kernel<<<1, 1024, 0, stream>>>(cnt, rowptr, N);
    csr_fill_kernel<<<edgeBlocks, thr, 0, stream>>>(eds, rowptr, cursor, eid, E);

    // ---- layer 1: relu([x | leaky(mean_w1(x))] @ W1 + b1) -> hA
    spmm_pull_kernel<<<nodeWaveBlocks, thr, 0, stream>>>(x, rowptr, eid, esr, ew1, act, N);
    gemm_wmma_kernel<2><<<gemmGrid, thr, 0, stream>>>(x, act, W1, b1, hA, N, 1);

    // ---- layer 2: relu([hA | leaky(mean_w2(hA))] @ W2 + b2) -> hB
    spmm_pull_kernel<<<nodeWaveBlocks, thr, 0, stream>>>(hA, rowptr, eid, esr, ew2, act, N);
    gemm_wmma_kernel<2><<<gemmGrid, thr, 0, stream>>>(hA, act, W2, b2, hB, N, 1);

    // ---- head: s = relu(hB @ Wl1 + bl1) -> act; out = s @ Wl2 + bl2
    gemm_wmma_kernel<1><<<gemmGrid, thr, 0, stream>>>(hB, nullptr, Wl1, bl1, act, N, 1);
    classifier_kernel<<<nodeWaveBlocks, thr, 0, stream>>>(act, Wl2, bl2, out, N);
}